// Generator_57071525429828
// MI455X (gfx1250) — compile-verified
//
#include <hip/hip_runtime.h>
#include <hip/hip_bf16.h>

typedef __attribute__((ext_vector_type(2))) float v2f;
typedef __attribute__((ext_vector_type(8))) float v8f;

#define COLOR_SIZE 8
#define F 128
#define L 196608            // 256*256*3
#define B 4
#define NCHUNK 768
#define CHUNK 256           // L / NCHUNK
#define SLD 130             // LDS row stride for S tile (even -> b64-aligned rows)

// ---------------- Pass 1: per-chunk feature sums ----------------
__global__ void pass1_partial_sums(const int* __restrict__ ex,
                                   const float* __restrict__ W_in,
                                   const float* __restrict__ b_in,
                                   const float* __restrict__ pos,
                                   float* __restrict__ partials) {
  const int blk = blockIdx.x;
  const int b   = blk / NCHUNK;
  const int ch  = blk % NCHUNK;
  const int f   = threadIdx.x;

  __shared__ float xs[CHUNK];
  const long l0 = (long)ch * CHUNK;
  const int* e = ex + (size_t)b * L + l0;
  for (int i = f; i < CHUNK; i += F)
    xs[i] = (float)e[i] * 0.5f - 3.0f;   // (e/8*2-1)*2-1
  __syncthreads();

  const float w  = W_in[f];
  const float bi = b_in[f];
  const float* pp = pos + (size_t)l0 * F + f;
  float sum = 0.0f;
#pragma unroll 4
  for (int i = 0; i < CHUNK; ++i) {
    float x = fmaf(xs[i], w, bi + pp[(size_t)i * F]);
    sum += fmaxf(x, 0.0f);
  }
  partials[((size_t)b * NCHUNK + ch) * F + f] = sum;
}

// ---------------- Pass 2: exclusive prefix over chunk sums ----------------
__global__ void pass2_scan_chunks(float* __restrict__ partials) {
  const int b = blockIdx.x;
  const int f = threadIdx.x;
  float run = 0.0f;
  float* p = partials + (size_t)b * NCHUNK * F + f;
  for (int c = 0; c < NCHUNK; c += 4) {
    float t0 = p[(size_t)(c + 0) * F];
    float t1 = p[(size_t)(c + 1) * F];
    float t2 = p[(size_t)(c + 2) * F];
    float t3 = p[(size_t)(c + 3) * F];
    p[(size_t)(c + 0) * F] = run; run += t0;
    p[(size_t)(c + 1) * F] = run; run += t1;
    p[(size_t)(c + 2) * F] = run; run += t2;
    p[(size_t)(c + 3) * F] = run; run += t3;
  }
}

// ---------------- Pass 3: cumsum + normalize + WMMA GEMM ----------------
__global__ void pass3_main(const int* __restrict__ ex,
                           const float* __restrict__ W_in,
                           const float* __restrict__ b_in,
                           const float* __restrict__ pos,
                           const float* __restrict__ W_final,
                           const float* __restrict__ b_final,
                           const float* __restrict__ prefix,
                           float* __restrict__ out) {
  const int blk  = blockIdx.x;
  const int b    = blk / NCHUNK;
  const int ch   = blk % NCHUNK;
  const int f    = threadIdx.x;       // feature owned by this thread
  const int lane = f & 31;
  const int wave = f >> 5;            // wave32: 4 waves per block
  const int hi   = (lane >= 16) ? 1 : 0;
  const int n    = lane & 15;         // M for A-frag, N for B/D-frags

  __shared__ float xs[CHUNK];
  __shared__ float S[16 * SLD];       // 16 positions x 128 features (+pad)
  __shared__ float Dw[4 * 256];       // per-wave 16x16 partial D
  __shared__ float SQ[16 * 8];        // per-row sum-of-squares partials

  const long l0 = (long)ch * CHUNK;
  const int* e = ex + (size_t)b * L + l0;
  for (int i = f; i < CHUNK; i += F)
    xs[i] = (float)e[i] * 0.5f - 3.0f;

  // B fragments: this wave covers K = wave*32 .. wave*32+31 in 8 steps of 4.
  // Layout (mirrors f32 A-frag): VGPR j, lanes 0-15: K=k0+j; lanes 16-31: K=k0+2+j.
  const int kbase = wave * 32 + hi * 2;
  v2f bfr[8];
#pragma unroll
  for (int j = 0; j < 8; ++j) {
    const int k = kbase + 4 * j;
    float w0 = (n < COLOR_SIZE) ? W_final[(size_t)k       * COLOR_SIZE + n] : 0.0f;
    float w1 = (n < COLOR_SIZE) ? W_final[(size_t)(k + 1) * COLOR_SIZE + n] : 0.0f;
    bfr[j].x = w0;
    bfr[j].y = w1;
  }

  const float w_in = W_in[f];
  const float bi   = b_in[f];
  float s = prefix[((size_t)b * NCHUNK + ch) * F + f];  // exclusive chunk prefix
  const float bf_c = b_final[f & 7];
  __syncthreads();

  const float* pp = pos + (size_t)l0 * F + f;

  for (int t = 0; t < CHUNK / 16; ++t) {
    // serial in-chunk cumsum for this feature; stage 16-row tile of s in LDS
#pragma unroll
    for (int i = 0; i < 16; ++i) {
      const int li = t * 16 + i;
      float x = fmaf(xs[li], w_in, bi + pp[(size_t)li * F]);
      s += fmaxf(x, 0.0f);
      S[i * SLD + f] = s;
    }
    __syncthreads();

    // D_partial(16x16) = S_tile(16 x 32-slice) x Wpad(32-slice x 16), f32 WMMA
    v8f acc = {};
#pragma unroll
    for (int j = 0; j < 8; ++j) {
      const int k = kbase + 4 * j;
      v2f a;
      a.x = S[n * SLD + k];
      a.y = S[n * SLD + k + 1];
      acc = __builtin_amdgcn_wmma_f32_16x16x4_f32(
          false, a, false, bfr[j], (short)0, acc, false, false);
    }
#pragma unroll
    for (int i = 0; i < 8; ++i) {
      const int row = hi ? (i + 8) : i;
      Dw[wave * 256 + row * 16 + n] = acc[i];
    }

    // sum-of-squares partials: thread -> (row = f>>3, 16-feature segment = f&7)
    {
      const int row = f >> 3, seg = f & 7;
      const float* sr = &S[row * SLD + seg * 16];
      float q = 0.0f;
#pragma unroll
      for (int k2 = 0; k2 < 16; ++k2) q = fmaf(sr[k2], sr[k2], q);
      SQ[row * 8 + seg] = q;
    }
    __syncthreads();

    // output: thread -> (row = f>>3, color c = f&7)
    {
      const int row = f >> 3, c = f & 7;
      float q = 0.0f;
#pragma unroll
      for (int k2 = 0; k2 < 8; ++k2) q += SQ[row * 8 + k2];
      q = fmaxf(q, 1e-12f);
      const float scale = 1.0f / sqrtf(q);
      const int di = row * 16 + c;
      const float d = Dw[di] + Dw[256 + di] + Dw[512 + di] + Dw[768 + di];
      out[((size_t)b * L + l0 + t * 16 + row) * COLOR_SIZE + c] = fmaf(d, scale, bf_c);
    }
    __syncthreads();
  }
}

extern "C" void kernel_launch(void* const* d_in, const int* in_sizes, int n_in,
                              void* d_out, int out_size, void* d_ws, size_t ws_size,
                              hipStream_t stream) {
  const int*   example = (const int*)  d_in[0];
  const float* W_in    = (const float*)d_in[1];
  const float* b_in    = (const float*)d_in[2];
  const float* pos     = (const float*)d_in[3];
  const float* W_final = (const float*)d_in[4];
  const float* b_final = (const float*)d_in[5];
  float* out      = (float*)d_out;
  float* partials = (float*)d_ws;   // B*NCHUNK*F floats = 1.5 MB

  pass1_partial_sums<<<B * NCHUNK, F, 0, stream>>>(example, W_in, b_in, pos, partials);
  pass2_scan_chunks<<<B, F, 0, stream>>>(partials);
  pass3_main<<<B * NCHUNK, F, 0, stream>>>(example, W_in, b_in, pos, W_final,
                                           b_final, partials, out);
}